// HungarianMatcher_55405078119158
// MI455X (gfx1250) — compile-verified
//
#include <hip/hip_runtime.h>
#include <hip/hip_bf16.h>

#define BS 64
#define NQ 900
#define NQ_PAD 912   // 57 * 16
#define NC 92        // multiple of 4 -> clean K loop for 16x16x4
#define NT 96
#define MTILES 57
#define NTILES 6

typedef __attribute__((ext_vector_type(2))) float v2f;
typedef __attribute__((ext_vector_type(8))) float v8f;

// ---------------------------------------------------------------------------
// Kernel 1: row softmax over logits -> prob, padded to NQ_PAD rows (zeros)
// ---------------------------------------------------------------------------
__global__ void hm_softmax_kernel(const float* __restrict__ logits,
                                  float* __restrict__ prob) {
    const int q   = blockIdx.x;   // 0..NQ_PAD-1
    const int b   = blockIdx.y;
    const int tid = threadIdx.x;  // 128
    __shared__ float red[128];

    float x = -3.4e38f;
    if (q < NQ && tid < NC)
        x = logits[((size_t)b * NQ + q) * NC + tid];
    red[tid] = x;
    __syncthreads();
    for (int s = 64; s > 0; s >>= 1) {
        if (tid < s) red[tid] = fmaxf(red[tid], red[tid + s]);
        __syncthreads();
    }
    const float m = red[0];
    __syncthreads();

    float e = 0.0f;
    if (q < NQ && tid < NC) e = expf(x - m);
    red[tid] = e;
    __syncthreads();
    for (int s = 64; s > 0; s >>= 1) {
        if (tid < s) red[tid] += red[tid + s];
        __syncthreads();
    }
    const float sum = red[0];

    if (tid < NC)
        prob[((size_t)b * NQ_PAD + q) * NC + tid] = (q < NQ) ? (e / sum) : 0.0f;
}

// ---------------------------------------------------------------------------
// Kernel 2: build one-hot(labels) B matrix: oh[b][c][t] = (label[b][t]==c)
// ---------------------------------------------------------------------------
__global__ void hm_onehot_kernel(const int* __restrict__ labels,
                                 float* __restrict__ oh) {
    const int b   = blockIdx.x;
    const int tid = threadIdx.x;  // 256
    float* o = oh + (size_t)b * NC * NT;
    for (int i = tid; i < NC * NT; i += 256) o[i] = 0.0f;
    __syncthreads();
    if (tid < NT) {
        const int c = labels[b * NT + tid];
        if (c >= 0 && c < NC) o[c * NT + tid] = 1.0f;
    }
}

// ---------------------------------------------------------------------------
// Kernel 3: fused cost tile. class cost via WMMA f32 16x16x4 (exact: one-hot
// GEMM gather), bbox L1 + GIoU in VALU epilogue straight from the accumulator.
// One wave32 per 16x16 (q,t) tile. EXEC is all-ones through the WMMA loop.
// ---------------------------------------------------------------------------
__global__ void hm_cost_kernel(const float* __restrict__ prob,
                               const float* __restrict__ oh,
                               const float* __restrict__ pred_boxes,
                               const float* __restrict__ tgt_boxes,
                               float* __restrict__ cost) {
    const int tm   = blockIdx.x;       // 0..56
    const int tn   = blockIdx.y;       // 0..5
    const int b    = blockIdx.z;
    const int lane = threadIdx.x;      // 32
    const int half = lane >> 4;        // 0: K=0,1  1: K=2,3 (A frag) / rows M or M+8 (C)
    const int l15  = lane & 15;

    // A: prob row for M = l15 within this tile (16x4 f32 A-frag layout)
    const float* A  = prob + ((size_t)b * NQ_PAD + (size_t)(tm * 16 + l15)) * NC;
    // B: one-hot, column n = l15 within this tile (4x16 f32 B-frag layout)
    const float* Bm = oh + (size_t)b * NC * NT + (tn * 16 + l15);

    v8f acc = {};
#pragma unroll
    for (int k = 0; k < NC; k += 4) {
        v2f a, bf;
        a.x  = A[k + (half ? 2 : 0)];
        a.y  = A[k + (half ? 3 : 1)];
        bf.x = Bm[(k + (half ? 2 : 0)) * NT];
        bf.y = Bm[(k + (half ? 3 : 1)) * NT];
        acc = __builtin_amdgcn_wmma_f32_16x16x4_f32(
            /*neg_a=*/false, a, /*neg_b=*/false, bf,
            /*c_mod=*/(short)0, acc, /*reuse_a=*/false, /*reuse_b=*/false);
    }

    // Epilogue: lane owns column t = tn*16 + l15; VGPR r owns row q = tm*16 + r + half*8
    const int t = tn * 16 + l15;
    const float* tb = tgt_boxes + ((size_t)b * NT + t) * 4;
    const float t0 = tb[0], t1 = tb[1], t2 = tb[2], t3 = tb[3];
    const float ta = (t2 - t0) * (t3 - t1);

#pragma unroll
    for (int r = 0; r < 8; ++r) {
        const int q = tm * 16 + r + half * 8;
        if (q >= NQ) continue;
        const float* pb = pred_boxes + ((size_t)b * NQ + q) * 4;
        const float p0 = pb[0], p1 = pb[1], p2 = pb[2], p3 = pb[3];

        const float cb = fabsf(p0 - t0) + fabsf(p1 - t1) +
                         fabsf(p2 - t2) + fabsf(p3 - t3);

        const float pa  = (p2 - p0) * (p3 - p1);
        const float iw  = fmaxf(fminf(p2, t2) - fmaxf(p0, t0), 0.0f);
        const float ih  = fmaxf(fminf(p3, t3) - fmaxf(p1, t1), 0.0f);
        const float inter = iw * ih;
        const float uni   = pa + ta - inter;
        const float iou   = inter / uni;
        const float ew  = fmaxf(fmaxf(p2, t2) - fminf(p0, t0), 0.0f);
        const float eh  = fmaxf(fmaxf(p3, t3) - fminf(p1, t1), 0.0f);
        const float enc = ew * eh;
        const float giou = iou - (enc - uni) / enc;

        const float cc = -acc[r];   // CLASS_COST * cost_class, exact gather
        cost[((size_t)b * NQ + q) * NT + t] = cc + 5.0f * cb + 2.0f * (-giou);
    }
}

// ---------------------------------------------------------------------------
// Kernel 4: per-batch matcher replicating the reference _lap EXACTLY
// (its minv/way copies are never written back, so each step is a fresh
//  argmin over free columns of C[i0-1][j-1] - u[i0] - v[j], with potential
//  updates, and the augmentation collapses to p[j_final] = i).
// One 1024-thread block per batch; argmin via LDS tree reduction; f64 math.
// ---------------------------------------------------------------------------
__global__ void hm_lap_kernel(const float* __restrict__ cost,
                              float* __restrict__ out_rows,
                              float* __restrict__ out_cols) {
    const int b   = blockIdx.x;
    const int tid = threadIdx.x;  // 1024

    __shared__ double v_[NQ + 1];
    __shared__ double u_[NT + 1];
    __shared__ int    p_[NQ + 1];
    __shared__ unsigned char used_[NQ + 1];
    __shared__ double redv[1024];
    __shared__ int    redi[1024];
    __shared__ int    s_j0;

    const float* C = cost + (size_t)b * NQ * NT;  // C.T row i0-1 = column t of cost

    for (int j = tid; j <= NQ; j += 1024) { v_[j] = 0.0; p_[j] = 0; }
    if (tid <= NT) u_[tid] = 0.0;
    __syncthreads();

    for (int i = 1; i <= NT; ++i) {
        if (tid == 0) { p_[0] = i; s_j0 = 0; }
        for (int j = tid; j <= NQ; j += 1024) used_[j] = 0;
        __syncthreads();

        while (true) {
            const int j0 = s_j0;
            if (tid == 0) used_[j0] = 1;
            __syncthreads();
            const int i0 = p_[j0];

            // candidate reduced cost for free column j = tid+1
            double val = 1e18;
            int    idx = 0;
            if (tid < NQ) {
                const int j = tid + 1;
                if (!used_[j]) {
                    val = (double)C[(size_t)tid * NT + (i0 - 1)] - u_[i0] - v_[j];
                    idx = j;
                }
            }
            redv[tid] = val; redi[tid] = idx;
            __syncthreads();
            for (int s = 512; s > 0; s >>= 1) {
                if (tid < s) {
                    const double ov = redv[tid + s];
                    const int    oi = redi[tid + s];
                    if (ov < redv[tid] ||
                        (ov == redv[tid] && oi != 0 &&
                         (redi[tid] == 0 || oi < redi[tid]))) {
                        redv[tid] = ov; redi[tid] = oi;
                    }
                }
                __syncthreads();
            }
            const double delta = redv[0];
            const int    j1    = redi[0];

            // u[p[used]] += delta ; v[used] -= delta  (rows in p over used cols
            // are distinct, so no write conflicts)
            if (tid <= NQ && used_[tid]) {
                u_[p_[tid]] += delta;
                v_[tid]     -= delta;
            }
            if (tid == 0) s_j0 = j1;
            __syncthreads();
            if (p_[j1] == 0) break;   // uniform decision across block
        }
        if (tid == 0) p_[s_j0] = i;   // way[] is all zeros in the reference
        __syncthreads();
    }

    if (tid == 0) {
        int k = 0;
        for (int j = 1; j <= NQ && k < NT; ++j) {
            if (p_[j] > 0) {
                out_rows[b * NT + k] = (float)(j - 1);
                out_cols[b * NT + k] = (float)(p_[j] - 1);
                ++k;
            }
        }
    }
}

// ---------------------------------------------------------------------------
extern "C" void kernel_launch(void* const* d_in, const int* in_sizes, int n_in,
                              void* d_out, int out_size, void* d_ws, size_t ws_size,
                              hipStream_t stream) {
    const float* pred_logits = (const float*)d_in[0];
    const float* pred_boxes  = (const float*)d_in[1];
    const int*   tgt_labels  = (const int*)d_in[2];
    const float* tgt_boxes   = (const float*)d_in[3];

    float* out      = (float*)d_out;
    float* cost     = out;                              // BS*NQ*NT
    float* out_rows = out + (size_t)BS * NQ * NT;       // BS*NT
    float* out_cols = out_rows + (size_t)BS * NT;       // BS*NT

    float* prob = (float*)d_ws;                         // BS*NQ_PAD*NC
    float* oh   = prob + (size_t)BS * NQ_PAD * NC;      // BS*NC*NT

    hm_softmax_kernel<<<dim3(NQ_PAD, BS), 128, 0, stream>>>(pred_logits, prob);
    hm_onehot_kernel<<<BS, 256, 0, stream>>>(tgt_labels, oh);
    hm_cost_kernel<<<dim3(MTILES, NTILES, BS), 32, 0, stream>>>(
        prob, oh, pred_boxes, tgt_boxes, cost);
    hm_lap_kernel<<<BS, 1024, 0, stream>>>(cost, out_rows, out_cols);
}